// MKGCN_28467043238497
// MI455X (gfx1250) — compile-verified
//
#include <hip/hip_runtime.h>
#include <hip/hip_bf16.h>
#include <math.h>

typedef __attribute__((ext_vector_type(2))) float v2f;
typedef __attribute__((ext_vector_type(8))) float v8f;

#define DIMX 32
#define KN 16

__device__ __forceinline__ float sig_(float x) { return 1.0f / (1.0f + __expf(-x)); }

__global__ __launch_bounds__(32) void kgcn_fused_kernel(
    const int* __restrict__ users, const int* __restrict__ items,
    const float* __restrict__ entity_emb, const float* __restrict__ relation_emb,
    const int* __restrict__ adj_entity, const int* __restrict__ adj_relation,
    const int* __restrict__ user_history,
    const float* __restrict__ Wm, const float* __restrict__ bias,
    float* __restrict__ out, int batch)
{
    __shared__ float s_user[DIMX];        // user embedding
    __shared__ float s_ent1[KN * DIMX];   // original hop-1 entity rows
    __shared__ float s_new1[KN * DIMX];   // updated hop-1 vectors (after it0,h1)
    __shared__ float s_X[KN * DIMX];      // matmul input staging
    __shared__ float s_scores[KN * KN];   // hop-2 attention logits
    __shared__ float s_sc1[KN];           // hop-1 attention logits
    __shared__ float s_vec[DIMX];         // 1x32 vector staging
    __shared__ int   s_e1[KN];            // hop-1 entity ids

    const int b = blockIdx.x;
    if (b >= batch) return;
    const int t      = threadIdx.x;   // 0..31 (wave32)
    const int lane16 = t & 15;
    const int half   = t >> 4;        // 0 or 1
    const int node   = t >> 1;        // 0..15 (lane pair -> node)
    const int sub    = t & 1;         // which half of the 16 neighbors

    // ---- user embedding: mean over 16 gathered history rows (lane = dim) ----
    const int uid = users[b];
    const long hbase = (long)uid * KN;
    float u_t = 0.0f;
    for (int k = 0; k < KN; ++k) {
        const int hid = user_history[hbase + k];
        u_t += entity_emb[(long)hid * DIMX + t];   // coalesced 128B per k
    }
    s_user[t] = u_t * (1.0f / 16.0f);

    // ---- hop-1 adjacency: e1 (lanes 0-15), r1 (lanes 16-31) ----
    const int item = items[b];
    int r1_t = 0;
    if (t < KN) s_e1[t] = adj_entity[(long)item * KN + t];
    else        r1_t    = adj_relation[(long)item * KN + lane16];
    __syncthreads();

    // user emb into registers (LDS broadcast)
    float u[DIMX];
#pragma unroll
    for (int i = 0; i < DIMX; ++i) u[i] = s_user[i];

    // ---- gather original ent1 rows -> LDS (each lane moves 16 contiguous floats) ----
#pragma unroll
    for (int q = 0; q < 4; ++q) {
        const int f   = t * 16 + q * 4;
        const int row = f >> 5, col = f & 31;
        const int eid = s_e1[row];
        const float4 v = *(const float4*)(entity_emb + (long)eid * DIMX + col);
        *(float4*)(s_ent1 + f) = v;
    }

    // ---- hop-1 attention logits (lanes 16-31): dot(u, relation_emb[r1]) ----
    if (t >= KN) {
        const float* rr = relation_emb + (long)r1_t * DIMX;
        float s = 0.0f;
#pragma unroll
        for (int q = 0; q < 8; ++q) {
            const float4 rv = *(const float4*)(rr + q * 4);
            s += u[q*4+0]*rv.x + u[q*4+1]*rv.y + u[q*4+2]*rv.z + u[q*4+3]*rv.w;
        }
        s_sc1[lane16] = s;
    }

    // ---- hop-2: lane pair per node; each lane handles 8 neighbors ----
    const int  e1n   = s_e1[node];
    const long abase = (long)e1n * KN + sub * 8;
    int eid2[8];
#pragma unroll
    for (int j = 0; j < 8; ++j) {
        eid2[j] = adj_entity[abase + j];
        const int rid = adj_relation[abase + j];
        __builtin_prefetch(entity_emb + (long)eid2[j] * DIMX, 0, 0);  // global_prefetch_b8
        const float* rr = relation_emb + (long)rid * DIMX;
        float s = 0.0f;
#pragma unroll
        for (int q = 0; q < 8; ++q) {
            const float4 rv = *(const float4*)(rr + q * 4);
            s += u[q*4+0]*rv.x + u[q*4+1]*rv.y + u[q*4+2]*rv.z + u[q*4+3]*rv.w;
        }
        s_scores[node * KN + sub * 8 + j] = s;
    }
    __syncthreads();

    // softmax over this node's 16 neighbors, keep own 8 weights
    float mx = -1e30f;
#pragma unroll
    for (int k = 0; k < KN; ++k) mx = fmaxf(mx, s_scores[node * KN + k]);
    float ssum = 0.0f;
#pragma unroll
    for (int k = 0; k < KN; ++k) ssum += __expf(s_scores[node * KN + k] - mx);
    const float sinv = 1.0f / ssum;
    float w8[8];
#pragma unroll
    for (int j = 0; j < 8; ++j)
        w8[j] = __expf(s_scores[node * KN + sub * 8 + j] - mx) * sinv;

    // attention-weighted neighbor sum (re-gather: L2-hot rows)
    float acc[DIMX];
#pragma unroll
    for (int i = 0; i < DIMX; ++i) acc[i] = 0.0f;
#pragma unroll
    for (int j = 0; j < 8; ++j) {
        const float* er = entity_emb + (long)eid2[j] * DIMX;
        const float w = w8[j];
#pragma unroll
        for (int q = 0; q < 8; ++q) {
            const float4 ev = *(const float4*)(er + q * 4);
            acc[q*4+0] += w * ev.x; acc[q*4+1] += w * ev.y;
            acc[q*4+2] += w * ev.z; acc[q*4+3] += w * ev.w;
        }
    }
#pragma unroll
    for (int i = 0; i < DIMX; ++i) acc[i] += __shfl_xor(acc[i], 1, 32);

    // X = old_ent1 + agg  (each lane writes its 16-col half of its node row)
#pragma unroll
    for (int i = 0; i < 16; ++i) {
        const int c = sub * 16 + i;
        s_X[node * DIMX + c] = s_ent1[node * DIMX + c] + acc[c];
    }
    __syncthreads();

    // ---- WMMA: H = sigmoid(X[16x32] @ W[32x32] + b) via v_wmma_f32_16x16x4_f32 ----
    v8f d0 = {}; v8f d1 = {};
#pragma unroll
    for (int kb = 0; kb < 8; ++kb) {
        const int ks = kb * 4 + 2 * half;
        v2f a;                                   // A 16x4: lane=row, VGPR0/1 = K {2h,2h+1}
        a.x = s_X[lane16 * DIMX + ks];
        a.y = s_X[lane16 * DIMX + ks + 1];
        v2f b0, b1;                              // B 4x16: lane=col, VGPR0/1 = K {2h,2h+1}
        b0.x = Wm[ks * DIMX + lane16];
        b0.y = Wm[(ks + 1) * DIMX + lane16];
        b1.x = Wm[ks * DIMX + 16 + lane16];
        b1.y = Wm[(ks + 1) * DIMX + 16 + lane16];
        d0 = __builtin_amdgcn_wmma_f32_16x16x4_f32(false, a, false, b0, (short)0, d0, false, false);
        d1 = __builtin_amdgcn_wmma_f32_16x16x4_f32(false, a, false, b1, (short)0, d1, false, false);
    }
    const float bc0 = bias[lane16];
    const float bc1 = bias[16 + lane16];
#pragma unroll
    for (int i = 0; i < 8; ++i) {               // D: VGPR i -> row i+8h, col = lane16
        const int row = i + 8 * half;
        s_new1[row * DIMX + lane16]      = sig_(d0[i] + bc0);
        s_new1[row * DIMX + 16 + lane16] = sig_(d1[i] + bc1);
    }
    __syncthreads();

    // ---- hop-0 attention weights (same for both iterations) ----
    float mx1 = -1e30f;
#pragma unroll
    for (int k = 0; k < KN; ++k) mx1 = fmaxf(mx1, s_sc1[k]);
    float a1[KN]; float sum1 = 0.0f;
#pragma unroll
    for (int k = 0; k < KN; ++k) { a1[k] = __expf(s_sc1[k] - mx1); sum1 += a1[k]; }
    const float inv1 = 1.0f / sum1;
#pragma unroll
    for (int k = 0; k < KN; ++k) a1[k] *= inv1;

    // ---- it0,h0: e0' = sigmoid((ent0 + sum_k a1[k]*old_ent1[k]) @ W + b) ----
    float agg0 = 0.0f;
#pragma unroll
    for (int k = 0; k < KN; ++k) agg0 += a1[k] * s_ent1[k * DIMX + t];
    const float ent0 = entity_emb[(long)item * DIMX + t];
    s_vec[t] = ent0 + agg0;
    __syncthreads();
    float y = bias[t];
#pragma unroll
    for (int i = 0; i < DIMX; ++i) y += s_vec[i] * Wm[i * DIMX + t];  // coalesced W rows
    const float e0a = sig_(y);
    __syncthreads();

    // ---- it1,h0: final = tanh((e0' + sum_k a1[k]*new_ent1[k]) @ W + b) ----
    float agg0b = 0.0f;
#pragma unroll
    for (int k = 0; k < KN; ++k) agg0b += a1[k] * s_new1[k * DIMX + t];
    s_vec[t] = e0a + agg0b;
    __syncthreads();
    float y2 = bias[t];
#pragma unroll
    for (int i = 0; i < DIMX; ++i) y2 += s_vec[i] * Wm[i * DIMX + t];
    const float fin = tanhf(y2);

    // ---- score = sigmoid(dot(user_emb, final)) ----
    float p = s_user[t] * fin;
#pragma unroll
    for (int off = 16; off >= 1; off >>= 1) p += __shfl_xor(p, off, 32);
    if (t == 0) out[b] = sig_(p);
}

extern "C" void kernel_launch(void* const* d_in, const int* in_sizes, int n_in,
                              void* d_out, int out_size, void* d_ws, size_t ws_size,
                              hipStream_t stream) {
    const int*   users        = (const int*)d_in[0];
    const int*   items        = (const int*)d_in[1];
    const float* entity_emb   = (const float*)d_in[2];
    const float* relation_emb = (const float*)d_in[3];
    const int*   adj_entity   = (const int*)d_in[4];
    const int*   adj_relation = (const int*)d_in[5];
    const int*   user_history = (const int*)d_in[6];
    const float* Wm           = (const float*)d_in[7];
    const float* bias         = (const float*)d_in[8];
    float*       out          = (float*)d_out;
    const int batch = in_sizes[0];

    hipLaunchKernelGGL(kgcn_fused_kernel, dim3(batch), dim3(32), 0, stream,
                       users, items, entity_emb, relation_emb, adj_entity,
                       adj_relation, user_history, Wm, bias, out, batch);
}